// MyModel_87522843560950
// MI455X (gfx1250) — compile-verified
//
#include <hip/hip_runtime.h>

typedef __attribute__((ext_vector_type(2))) float v2f;
typedef __attribute__((ext_vector_type(8))) float v8f;

#define NUM_ITERS 100
#define LR 0.1f

// Kernel 1: single-pass row partial sums.
// grid = (chunks, rows), 256 threads/block (8 wave32).
// Coalesced float4 loads; per-thread dual accumulators feed a
// V_WMMA_F32_16X16X4_F32 cross-lane reduction (B = ones).
__global__ __launch_bounds__(256) void row_partial_sums(
    const float* __restrict__ data, float* __restrict__ partials,
    int N, int chunks)
{
    const int row = blockIdx.y;
    const long long rowBase = (long long)row * (long long)N;
    const float4* __restrict__ v = (const float4*)(data + rowBase);
    const int N4 = N >> 2;

    float a0 = 0.0f, a1 = 0.0f;
    int i = blockIdx.x * blockDim.x + threadIdx.x;
    const int stride = gridDim.x * blockDim.x;
    for (; i < N4; i += stride) {
        float4 p = v[i];
        a0 += p.x + p.y;
        a1 += p.z + p.w;
    }
    // Tail (N % 4) handled by low threads of chunk 0 (N=500000 -> tail 0, but stay general).
    if (blockIdx.x == 0) {
        const int tail = N & 3;
        if ((int)threadIdx.x < tail)
            a0 += data[rowBase + (long long)(N4 << 2) + (long long)threadIdx.x];
    }
    // EXEC reconverges to all-1s here (required for WMMA).

    // Wave-level sum via WMMA: A (16x4 f32) holds the wave's 64 partials
    // (lane L, VGPR v <-> M=L%16, K=v+2*(L/16)); B = ones so
    // D[m][n] = sum_k A[m][k]. D layout: lane L, VGPR j = D[j+8*(L/16)][L%16].
    // Summing the 8 D regs gives each half-wave's total; one xor-16 shuffle
    // combines halves -> full 64-value wave sum in every lane.
    v2f A; A[0] = a0;   A[1] = a1;
    v2f B; B[0] = 1.0f; B[1] = 1.0f;
    v8f C = {0.f, 0.f, 0.f, 0.f, 0.f, 0.f, 0.f, 0.f};
    v8f D = __builtin_amdgcn_wmma_f32_16x16x4_f32(
        /*neg_a=*/false, A, /*neg_b=*/false, B,
        /*c_mod=*/(short)0, C, /*reuse_a=*/false, /*reuse_b=*/false);
    float t = ((D[0] + D[1]) + (D[2] + D[3])) + ((D[4] + D[5]) + (D[6] + D[7]));
    t += __shfl_xor(t, 16, 32);

    __shared__ float wsum[8];
    const int wave = threadIdx.x >> 5;
    if ((threadIdx.x & 31) == 0) wsum[wave] = t;
    __syncthreads();
    if (threadIdx.x == 0) {
        float s = ((wsum[0] + wsum[1]) + (wsum[2] + wsum[3]))
                + ((wsum[4] + wsum[5]) + (wsum[6] + wsum[7]));
        partials[(long long)row * chunks + blockIdx.x] = s;
    }
}

// Kernel 2: deterministic final reduce per row + the 100 SGD steps in registers.
// grad = mean(2*(x - data)) == 2*(x - mean(data)) by linearity of the mean.
__global__ void sgd_finish(const float* __restrict__ partials,
                           float* __restrict__ out,
                           int chunks, int rows, float invN)
{
    const int r = blockIdx.x * blockDim.x + threadIdx.x;
    if (r >= rows) return;
    float s = 0.0f;
    for (int c = 0; c < chunks; ++c)
        s += partials[(long long)r * chunks + c];
    const float m = s * invN;

    float x = 0.0f;
#pragma unroll 1
    for (int it = 0; it < NUM_ITERS; ++it) {
        const float grad = 2.0f * (x - m);
        x = x - LR * grad;
    }
    out[r] = x;
}

extern "C" void kernel_launch(void* const* d_in, const int* in_sizes, int n_in,
                              void* d_out, int out_size, void* d_ws, size_t ws_size,
                              hipStream_t stream) {
    const float* reps = (const float*)d_in[0];
    float* out       = (float*)d_out;
    float* partials  = (float*)d_ws;

    const int rows  = out_size;            // 64
    const int total = in_sizes[0];         // 64 * 500000
    const int N     = total / rows;        // 500000

    int chunks = 32;                       // 32 chunks * 64 rows = 2048 blocks
    while ((size_t)rows * (size_t)chunks * sizeof(float) > ws_size && chunks > 1)
        chunks >>= 1;                      // ws-size safety (needs only 8 KB)

    dim3 grid(chunks, rows);
    row_partial_sums<<<grid, 256, 0, stream>>>(reps, partials, N, chunks);

    const int tb = 64;
    sgd_finish<<<(rows + tb - 1) / tb, tb, 0, stream>>>(
        partials, out, chunks, rows, 1.0f / (float)N);
}